// FeatureExtractorMatchedFilterMaxDir_39530878992458
// MI455X (gfx1250) — compile-verified
//
#include <hip/hip_runtime.h>
#include <hip/hip_bf16.h>
#include <stdint.h>

// Problem dims (fixed by the reference)
#define B_  4
#define T_  1000
#define F_  257
#define C_  6
#define D_  2048
#define O_  2

static constexpr int   FC      = F_ * C_;           // 1542 floats per (b,t) per array
static constexpr int   FD      = F_ * D_;           // 526336
static constexpr float DEG2RAD = 0.017453292519943295f;

typedef float v4f __attribute__((ext_vector_type(4)));
typedef float v2f __attribute__((ext_vector_type(2)));

// ---------------------------------------------------------------------------
// K0: per-direction trig table {cos(azi), sin(azi), cos(zen), sin(zen)}.
// 2048 * 16B = 32 KB, lives in d_ws, L2-resident for the main kernel.
// ---------------------------------------------------------------------------
__global__ __launch_bounds__(256)
void trig_table_kernel(const float* __restrict__ dirs, v4f* __restrict__ tab)
{
    int d = blockIdx.x * blockDim.x + threadIdx.x;
    if (d < D_) {
        float da = dirs[2 * d];
        float dz = dirs[2 * d + 1];
        v4f t = {cosf(da), sinf(da), cosf(dz), sinf(dz)};
        tab[d] = t;
    }
}

// ---------------------------------------------------------------------------
// Main kernel: one 256-thread block per (b,t).
// ---------------------------------------------------------------------------
__global__ __launch_bounds__(256)
void mfmd_kernel(const float* __restrict__ Xr,  const float* __restrict__ Xi,
                 const float* __restrict__ tdoa,
                 const v4f*   __restrict__ tab,
                 const float* __restrict__ wcr, const float* __restrict__ wci,
                 const float* __restrict__ wdr, const float* __restrict__ wdi,
                 const float* __restrict__ wbr, const float* __restrict__ wbi,
                 float* __restrict__ out)
{
    const int bt  = blockIdx.x;     // 0 .. B*T-1
    const int tid = threadIdx.x;    // 0 .. 255

    __shared__ float sXr[FC];
    __shared__ float sXi[FC];
    __shared__ float ra[8];
    __shared__ int   ri[8];
    __shared__ int   sInd;

    // ------------------------------------------------------------------
    // Phase 0: async DMA of X[b,t,:,:] (12336 B) into LDS; overlaps with
    // the argmin below. 771 B64 chunks per array = 3*257.
    // ------------------------------------------------------------------
    {
        const float* gXr = Xr + (size_t)bt * FC;
        const float* gXi = Xi + (size_t)bt * FC;
        uint32_t lr = (uint32_t)(uintptr_t)&sXr[0];
        uint32_t li = (uint32_t)(uintptr_t)&sXi[0];
        int i = tid;
        #pragma unroll
        for (int k = 0; k < 3; ++k) {                 // covers i = 0..767
            asm volatile("global_load_async_to_lds_b64 %0, %1, off"
                         :: "v"(lr + 8u * (uint32_t)i), "v"(gXr + 2 * i) : "memory");
            asm volatile("global_load_async_to_lds_b64 %0, %1, off"
                         :: "v"(li + 8u * (uint32_t)i), "v"(gXi + 2 * i) : "memory");
            i += 256;
        }
        if (i < FC / 2) {                             // tail: lanes 0..2
            asm volatile("global_load_async_to_lds_b64 %0, %1, off"
                         :: "v"(lr + 8u * (uint32_t)i), "v"(gXr + 2 * i) : "memory");
            asm volatile("global_load_async_to_lds_b64 %0, %1, off"
                         :: "v"(li + 8u * (uint32_t)i), "v"(gXi + 2 * i) : "memory");
        }
    }

    // ------------------------------------------------------------------
    // Phase 1: haversine argmin over D dirs — trig-free inner loop via
    //   sin^2((x-y)/2) = 0.5*(1 - cos x cos y - sin x sin y)
    // angle = 2*asin(sqrt(clip(a,0,1))) is monotone in a -> argmin on a.
    // ------------------------------------------------------------------
    const float tazi = tdoa[2 * bt]     * DEG2RAD;
    const float tzen = tdoa[2 * bt + 1] * DEG2RAD;
    const float ca = cosf(tazi), sa = sinf(tazi);
    const float ct = cosf(tzen), st = sinf(tzen);

    float best = 1e30f;
    int   bidx = 0;
    for (int d = tid; d < D_; d += 256) {
        v4f t = tab[d];                 // {cos da, sin da, cos dz, sin dz}
        float havz = 0.5f - 0.5f * (ct * t.z + st * t.w);   // sin^2(zen_diff/2)
        float hava = 0.5f - 0.5f * (ca * t.x + sa * t.y);   // sin^2(azi_diff/2)
        float a = havz + ct * t.z * hava;
        if (a < best) { best = a; bidx = d; }   // strict <: first-min wins
    }

    // wave32 intra-wave argmin (tie-break: smaller global index)
    #pragma unroll
    for (int off = 16; off > 0; off >>= 1) {
        float o  = __shfl_down(best, off, 32);
        int   oi = __shfl_down(bidx, off, 32);
        if (o < best || (o == best && oi < bidx)) { best = o; bidx = oi; }
    }
    if ((tid & 31) == 0) { ra[tid >> 5] = best; ri[tid >> 5] = bidx; }
    __syncthreads();
    if (tid == 0) {
        float b = ra[0]; int bi0 = ri[0];
        #pragma unroll
        for (int w = 1; w < 8; ++w) {
            float o = ra[w]; int oi = ri[w];
            if (o < b || (o == b && oi < bi0)) { b = o; bi0 = oi; }
        }
        sInd = bi0;
    }

    // Wait for this wave's async LDS writes, then make everything
    // (staged X + sInd) block-visible.
    asm volatile("s_wait_asynccnt 0x0" ::: "memory");
    __syncthreads();
    const int ind = sInd;

    // ------------------------------------------------------------------
    // Phase 2: gather weight columns at `ind`, complex math, NT stores.
    //   out1[b,t,f,c,{re,im}] = w_ds[c,f,ind]   * X[b,t,f,c]
    //   out2[b,t,f,{re,im}]   = sum_c w_conj[c,f,ind] * X[b,t,f,c]
    //   out3[b,t,f,o,{re,im}] = w_bin[o,f,ind]
    // ------------------------------------------------------------------
    float* out1 = out;
    float* out2 = out1 + (size_t)B_ * T_ * F_ * C_ * 2;   // +12,336,000
    float* out3 = out2 + (size_t)B_ * T_ * F_ * 2;        // +2,056,000

    for (int f = tid; f < F_; f += 256) {
        const int wbase = f * D_ + ind;

        float br = 0.f, bi = 0.f;
        float o1[12];
        #pragma unroll
        for (int c = 0; c < C_; ++c) {
            float xr = sXr[f * C_ + c];
            float xi = sXi[f * C_ + c];
            int   wj = c * FD + wbase;

            float dr = wdr[wj], di = wdi[wj];
            o1[2 * c]     = dr * xr - di * xi;
            o1[2 * c + 1] = dr * xi + di * xr;

            float cr = wcr[wj], ci = wci[wj];
            br += cr * xr - ci * xi;
            bi += cr * xi + ci * xr;
        }

        {   // 48 B per f, 16B-aligned -> 3x b128 NT streaming stores
            v4f* p = (v4f*)(out1 + ((size_t)bt * F_ + f) * (C_ * 2));
            v4f a0 = {o1[0], o1[1], o1[2],  o1[3]};
            v4f a1 = {o1[4], o1[5], o1[6],  o1[7]};
            v4f a2 = {o1[8], o1[9], o1[10], o1[11]};
            __builtin_nontemporal_store(a0, p);
            __builtin_nontemporal_store(a1, p + 1);
            __builtin_nontemporal_store(a2, p + 2);
        }
        {   // bf_out: b64 NT store
            v2f* p = (v2f*)(out2 + ((size_t)bt * F_ + f) * 2);
            v2f v = {br, bi};
            __builtin_nontemporal_store(v, p);
        }
        {   // binaural gather: 4 floats, 16B-aligned -> b128 NT store
            float b0r = wbr[wbase],      b0i = wbi[wbase];
            float b1r = wbr[FD + wbase], b1i = wbi[FD + wbase];
            v4f* p = (v4f*)(out3 + ((size_t)bt * F_ + f) * 4);
            v4f v = {b0r, b0i, b1r, b1i};
            __builtin_nontemporal_store(v, p);
        }
    }
}

extern "C" void kernel_launch(void* const* d_in, const int* in_sizes, int n_in,
                              void* d_out, int out_size, void* d_ws, size_t ws_size,
                              hipStream_t stream) {
    (void)in_sizes; (void)n_in; (void)ws_size; (void)out_size;

    const float* Xr   = (const float*)d_in[0];
    const float* Xi   = (const float*)d_in[1];
    const float* tdoa = (const float*)d_in[2];
    const float* dirs = (const float*)d_in[3];
    const float* wcr  = (const float*)d_in[4];
    const float* wci  = (const float*)d_in[5];
    const float* wdr  = (const float*)d_in[6];
    const float* wdi  = (const float*)d_in[7];
    const float* wbr  = (const float*)d_in[8];
    const float* wbi  = (const float*)d_in[9];

    v4f* tab = (v4f*)d_ws;   // 2048 * 16 B = 32 KB of scratch

    trig_table_kernel<<<dim3((D_ + 255) / 256), dim3(256), 0, stream>>>(dirs, tab);

    mfmd_kernel<<<dim3(B_ * T_), dim3(256), 0, stream>>>(
        Xr, Xi, tdoa, tab, wcr, wci, wdr, wdi, wbr, wbi, (float*)d_out);
}